// PaiNN_48825188221160
// MI455X (gfx1250) — compile-verified
//
#include <hip/hip_runtime.h>
#include <hip/hip_bf16.h>
#include <math.h>

// ---------------- problem constants (from reference) ----------------
#define NN    8192        // atoms
#define EE    122880      // edges
#define FF    128         // feature width
#define RR    64          // rbf width
#define LL    3           // blocks
#define CUTOFF 5.0f

typedef float v2f __attribute__((ext_vector_type(2)));
typedef float v8f __attribute__((ext_vector_type(8)));

// A-source modes for the fused MLP kernel
#define A_PLAIN   0
#define A_GATHER  1   // row r = concat( s[rowIdx[r]][0:128], rbf[r][0:64] )  (DIN=192)
#define A_CONCAT2 2   // row r = concat( A1[r][0:128], A2[r][0:128] )        (DIN=256)

// epilogue modes
#define E_STORE     0
#define E_SCATTER_S 1 // atomicAdd into OUT[colIdx[r]*128 + n]
#define E_SCATTER_V 2 // normalize v_diff and atomicAdd 3 planes into OUT[colIdx[r]*384 + d*128 + n]
#define E_RESIDUAL  3 // OUT[r*128+n] = RES[r*128+n] + val

__device__ __forceinline__ float silu_f(float x) {
    return x * (1.0f / (1.0f + __expf(-x)));
}

// ---------------------------------------------------------------------------
// Weights are pre-packed: for W (din x dout) row-major,
//   Wp[(k>>1)*(2*dout) + 2*n + (k&1)] = W[k*dout + n]
// so a WMMA B-fragment {W[ka][n], W[ka+1][n]} is ONE aligned b64 load.
// ---------------------------------------------------------------------------
__global__ void pack_w_kernel(const float* __restrict__ W,
                              float* __restrict__ Wp,
                              int din, int dout) {
    const int idx = blockIdx.x * 256 + threadIdx.x;
    if (idx >= din * dout) return;
    const int k = idx / dout;
    const int n = idx - k * dout;
    Wp[(size_t)(k >> 1) * (2 * dout) + 2 * n + (k & 1)] = W[idx];
}

// ---------------------------------------------------------------------------
// Fused (optionally 2-layer) MLP GEMM using V_WMMA_F32_16X16X4_F32.
// One workgroup = 256 threads = 8 waves; tile = 128 rows.
// Each wave owns 16 rows and all NT=8 column tiles (128 cols).
// A from LDS (ds_load_2addr_b64); B fragments are single b64 loads from
// packed weights, DOUBLE-BUFFERED across K-steps so WMMAs overlap loads.
// Hidden tile (silu) kept in LDS; Layer2 reads it back via ds_load.
// ---------------------------------------------------------------------------
template<int DIN, int AMODE, int EMODE, bool TWO_LAYER, bool SILU2>
__global__ void painn_mlp_kernel(
    const float* __restrict__ A1,     // plain/concat2 first half
    const float* __restrict__ A2,     // concat2 second half, or rbf for gather
    const float* __restrict__ SNODE,  // gather: node scalar features
    const int*   __restrict__ rowIdx, // gather / scatter_v
    const int*   __restrict__ colIdx, // scatter
    const float* __restrict__ W1p, const float* __restrict__ B1,
    const float* __restrict__ W2p, const float* __restrict__ B2,
    const float* __restrict__ VRES,   // scatter_v: v_cur [N,3,128]; residual: source
    float*       __restrict__ OUT,
    int dout1)                        // layer-1 output width
{
    constexpr int MT = 128;
    constexpr int NT = 8;             // 8 * 16 = 128 cols per wave per pass
    const int tile_m = blockIdx.x * MT;
    const int n0     = blockIdx.y * (NT * 16);   // only used when !TWO_LAYER

    extern __shared__ float lds[];
    float* Xs = lds;                               // [MT, DIN]
    float* Hs = lds + MT * DIN;                    // [MT, 128] if TWO_LAYER

    const int tid = threadIdx.x;

    // ---------------- stage A tile into LDS (fused gather/concat) ----------
    for (int i = tid; i < MT * DIN; i += 256) {
        const int r = i / DIN;
        const int c = i - r * DIN;
        const int gr = tile_m + r;
        float val;
        if (AMODE == A_PLAIN) {
            val = A1[(size_t)gr * DIN + c];
        } else if (AMODE == A_CONCAT2) {
            val = (c < 128) ? A1[(size_t)gr * 128 + c]
                            : A2[(size_t)gr * 128 + (c - 128)];
        } else { // A_GATHER
            if (c < 128) val = SNODE[(size_t)rowIdx[gr] * 128 + c];
            else         val = A2[(size_t)gr * 64 + (c - 128)];
        }
        Xs[i] = val;
    }
    __syncthreads();

    const int lane = tid & 31;
    const int wave = tid >> 5;
    const int hl   = lane >> 4;      // half-wave select
    const int l15  = lane & 15;
    const int mw   = wave * 16;      // wave's row base inside tile
    const int nb   = TWO_LAYER ? 0 : n0;

    // ---------------- layer 1 (double-buffered B) ----------------
    v8f acc[NT];
#pragma unroll
    for (int t = 0; t < NT; ++t) acc[t] = (v8f){0,0,0,0,0,0,0,0};

    const size_t w1stride = (size_t)(2 * dout1);
    v2f bf[NT];
#pragma unroll
    for (int t = 0; t < NT; ++t)
        bf[t] = *reinterpret_cast<const v2f*>(
            &W1p[(size_t)hl * w1stride + 2 * (nb + t * 16 + l15)]);

#pragma unroll 1
    for (int k = 0; k < DIN - 4; k += 4) {
        v2f a = *reinterpret_cast<const v2f*>(&Xs[(mw + l15) * DIN + k + 2 * hl]);
        v2f bn[NT];
#pragma unroll
        for (int t = 0; t < NT; ++t)
            bn[t] = *reinterpret_cast<const v2f*>(
                &W1p[(size_t)(((k + 4) >> 1) + hl) * w1stride + 2 * (nb + t * 16 + l15)]);
#pragma unroll
        for (int t = 0; t < NT; ++t)
            acc[t] = __builtin_amdgcn_wmma_f32_16x16x4_f32(
                false, a, false, bf[t], (short)0, acc[t], false, false);
#pragma unroll
        for (int t = 0; t < NT; ++t) bf[t] = bn[t];
    }
    {
        const int k = DIN - 4;
        v2f a = *reinterpret_cast<const v2f*>(&Xs[(mw + l15) * DIN + k + 2 * hl]);
#pragma unroll
        for (int t = 0; t < NT; ++t)
            acc[t] = __builtin_amdgcn_wmma_f32_16x16x4_f32(
                false, a, false, bf[t], (short)0, acc[t], false, false);
    }

    if constexpr (!TWO_LAYER) {
        // single linear layer (init_vector): bias, no activation, store
#pragma unroll
        for (int t = 0; t < NT; ++t) {
            const int n = n0 + t * 16 + l15;
            const float bias = B1[n];
#pragma unroll
            for (int j = 0; j < 8; ++j) {
                const int gm = tile_m + mw + j + 8 * hl;
                OUT[(size_t)gm * dout1 + n] = acc[t][j] + bias;
            }
        }
        return;
    } else {
        // bias + SiLU, write hidden tile to LDS
#pragma unroll
        for (int t = 0; t < NT; ++t) {
            const int n = t * 16 + l15;
            const float bias = B1[n];
#pragma unroll
            for (int j = 0; j < 8; ++j) {
                const int ml = mw + j + 8 * hl;
                Hs[ml * 128 + n] = silu_f(acc[t][j] + bias);
            }
        }
        __syncthreads();

        // ---------------- layer 2 (128 -> 128, double-buffered B) ----------
        v8f acc2[NT];
#pragma unroll
        for (int t = 0; t < NT; ++t) acc2[t] = (v8f){0,0,0,0,0,0,0,0};

        v2f bf2[NT];
#pragma unroll
        for (int t = 0; t < NT; ++t)
            bf2[t] = *reinterpret_cast<const v2f*>(
                &W2p[(size_t)hl * 256 + 2 * (t * 16 + l15)]);

#pragma unroll 1
        for (int k = 0; k < 128 - 4; k += 4) {
            v2f a = *reinterpret_cast<const v2f*>(&Hs[(mw + l15) * 128 + k + 2 * hl]);
            v2f bn[NT];
#pragma unroll
            for (int t = 0; t < NT; ++t)
                bn[t] = *reinterpret_cast<const v2f*>(
                    &W2p[(size_t)(((k + 4) >> 1) + hl) * 256 + 2 * (t * 16 + l15)]);
#pragma unroll
            for (int t = 0; t < NT; ++t)
                acc2[t] = __builtin_amdgcn_wmma_f32_16x16x4_f32(
                    false, a, false, bf2[t], (short)0, acc2[t], false, false);
#pragma unroll
            for (int t = 0; t < NT; ++t) bf2[t] = bn[t];
        }
        {
            v2f a = *reinterpret_cast<const v2f*>(&Hs[(mw + l15) * 128 + 124 + 2 * hl]);
#pragma unroll
            for (int t = 0; t < NT; ++t)
                acc2[t] = __builtin_amdgcn_wmma_f32_16x16x4_f32(
                    false, a, false, bf2[t], (short)0, acc2[t], false, false);
        }

        // ---------------- epilogue ----------------
#pragma unroll
        for (int t = 0; t < NT; ++t) {
            const int n = t * 16 + l15;
            const float bias = B2[n];
#pragma unroll
            for (int j = 0; j < 8; ++j) {
                const int gm = tile_m + mw + j + 8 * hl;
                float val = acc2[t][j] + bias;
                if (SILU2) val = silu_f(val);
                if (EMODE == E_STORE) {
                    OUT[(size_t)gm * 128 + n] = val;
                } else if (EMODE == E_RESIDUAL) {
                    OUT[(size_t)gm * 128 + n] = VRES[(size_t)gm * 128 + n] + val;
                } else if (EMODE == E_SCATTER_S) {
                    atomicAdd(&OUT[(size_t)colIdx[gm] * 128 + n], val);
                } else if (EMODE == E_SCATTER_V) {
                    const int r = rowIdx[gm];
                    const int c = colIdx[gm];
                    const float d0 = VRES[(size_t)r * 384 + n]       - VRES[(size_t)c * 384 + n];
                    const float d1 = VRES[(size_t)r * 384 + 128 + n] - VRES[(size_t)c * 384 + 128 + n];
                    const float d2 = VRES[(size_t)r * 384 + 256 + n] - VRES[(size_t)c * 384 + 256 + n];
                    const float inv = val / (sqrtf(d0*d0 + d1*d1 + d2*d2) + 1e-8f);
                    atomicAdd(&OUT[(size_t)c * 384 + n],       inv * d0);
                    atomicAdd(&OUT[(size_t)c * 384 + 128 + n], inv * d1);
                    atomicAdd(&OUT[(size_t)c * 384 + 256 + n], inv * d2);
                }
            }
        }
    }
}

// ---------------- small side kernels ----------------

__global__ void embed_kernel(const int* __restrict__ atoms,
                             const float* __restrict__ emb,
                             float* __restrict__ s) {
    const int idx = blockIdx.x * 256 + threadIdx.x;   // N*128 threads
    const int n = idx >> 7, f = idx & 127;
    s[idx] = emb[atoms[n] * 128 + f];
}

__global__ void rbf_kernel(const float* __restrict__ pos,
                           const int* __restrict__ ei,
                           float* __restrict__ rbf) {
    const int idx = blockIdx.x * 256 + threadIdx.x;   // E*64 threads
    const int e = idx >> 6, i = idx & 63;
    const int r = ei[e], c = ei[EE + e];
    const float dx = pos[r * 3 + 0] - pos[c * 3 + 0];
    const float dy = pos[r * 3 + 1] - pos[c * 3 + 1];
    const float dz = pos[r * 3 + 2] - pos[c * 3 + 2];
    const float d = sqrtf(dx * dx + dy * dy + dz * dz);
    const float sig = CUTOFF / 63.0f;
    const float mean = (float)i * sig;
    const float u = d - mean;
    const float val = __expf(-u * u / (2.0f * sig * sig));
    rbf[idx] = (d < CUTOFF) ? val : 0.0f;
}

__global__ void out3_kernel(const float* __restrict__ h,
                            const float* __restrict__ w3,
                            const float* __restrict__ b3,
                            float* __restrict__ out) {
    const int n = blockIdx.x * 256 + threadIdx.x;
    if (n >= NN) return;
    float acc = b3[0];
#pragma unroll 8
    for (int f = 0; f < 128; ++f) acc += h[(size_t)n * 128 + f] * w3[f];
    out[n] = acc;
}

// ---------------- host side ----------------
// Param flattening (JAX pytree, dicts by sorted key):
//   0 atoms, 1 atom_positions, 2 graph_indexes, 3 edge_index,
//   4 + 16*b + {0..15}:  msg_scalar{l1.b,l1.w,l2.b,l2.w} msg_vector{...}
//                        upd_scalar{...} upd_vector{...}
//   52 embedding, 53 init_vector.b, 54 init_vector.w,
//   55 out1.b, 56 out1.w, 57 out2.b, 58 out2.w, 59 out3.b, 60 out3.w

extern "C" void kernel_launch(void* const* d_in, const int* in_sizes, int n_in,
                              void* d_out, int out_size, void* d_ws, size_t ws_size,
                              hipStream_t stream) {
    auto F32 = [&](int i) { return (const float*)d_in[i]; };
    const int*   atoms = (const int*)d_in[0];
    const float* pos   = F32(1);
    const int*   ei    = (const int*)d_in[3];

    // workspace layout (floats)
    float* ws = (float*)d_ws;
    float* s0     = ws;                        // N*128
    float* s1     = s0 + (size_t)NN * 128;
    float* v0     = s1 + (size_t)NN * 128;     // N*384
    float* v1     = v0 + (size_t)NN * 384;
    float* ds_agg = v1 + (size_t)NN * 384;     // N*128
    float* dv_agg = ds_agg + (size_t)NN * 128; // N*384
    float* rbf    = dv_agg + (size_t)NN * 384; // E*64
    float* hhead  = rbf + (size_t)EE * 64;     // N*128
    float* wpcur  = hhead + (size_t)NN * 128;  // packed weights (~2.5MB)

    const int threads = 256;

    // ---- pack all weight matrices into b64-fragment-friendly layout ----
    auto pack = [&](int idx, int din, int dout) -> const float* {
        float* dst = wpcur;
        wpcur += (size_t)din * dout;
        const int total = din * dout;
        pack_w_kernel<<<(total + 255) / 256, threads, 0, stream>>>(
            F32(idx), dst, din, dout);
        return dst;
    };
    const float* wp_init = pack(54, 128, 384);
    const float* wp_b[LL][8];
    for (int b = 0; b < LL; ++b) {
        const int pb = 4 + 16 * b;
        wp_b[b][0] = pack(pb + 1, 192, 128);   // msg_scalar.l1.w
        wp_b[b][1] = pack(pb + 3, 128, 128);   // msg_scalar.l2.w
        wp_b[b][2] = pack(pb + 5, 192, 128);   // msg_vector.l1.w
        wp_b[b][3] = pack(pb + 7, 128, 128);   // msg_vector.l2.w
        wp_b[b][4] = pack(pb + 9, 256, 128);   // upd_scalar.l1.w
        wp_b[b][5] = pack(pb + 11, 128, 128);  // upd_scalar.l2.w
        wp_b[b][6] = pack(pb + 13, 256, 128);  // upd_vector.l1.w
        wp_b[b][7] = pack(pb + 15, 128, 128);  // upd_vector.l2.w
    }
    const float* wp_out1 = pack(56, 128, 128);
    const float* wp_out2 = pack(58, 128, 128);

    // s = embedding[atoms]
    embed_kernel<<<NN * 128 / threads, threads, 0, stream>>>(atoms, F32(52), s0);
    // rbf
    rbf_kernel<<<EE * 64 / threads, threads, 0, stream>>>(pos, ei, rbf);

    // v = (s @ W_init + b_init)  -> [N,3,128] ; single-layer, DIN=128, dout=384
    {
        dim3 grid(NN / 128, 3);
        size_t smem = (size_t)128 * 128 * 4;
        painn_mlp_kernel<128, A_PLAIN, E_STORE, false, false>
            <<<grid, threads, smem, stream>>>(
                s0, nullptr, nullptr, nullptr, nullptr,
                wp_init, F32(53), nullptr, nullptr, nullptr, v0, 384);
    }

    float* s_cur = s0; float* s_nxt = s1;
    float* v_cur = v0; float* v_nxt = v1;

    for (int b = 0; b < LL; ++b) {
        const int pb = 4 + 16 * b;
        hipMemsetAsync(ds_agg, 0, (size_t)NN * 128 * 4, stream);
        hipMemsetAsync(dv_agg, 0, (size_t)NN * 384 * 4, stream);

        // msg_scalar: [E,192] -> silu -> [E,128] -> [E,128], scatter-add by col
        {
            dim3 grid(EE / 128, 1);
            size_t smem = (size_t)(128 * 192 + 128 * 128) * 4;   // 160 KB
            painn_mlp_kernel<192, A_GATHER, E_SCATTER_S, true, false>
                <<<grid, threads, smem, stream>>>(
                    nullptr, rbf, s_cur, ei, ei + EE,
                    wp_b[b][0], F32(pb + 0), wp_b[b][1], F32(pb + 2),
                    nullptr, ds_agg, 128);
        }
        // msg_vector: same input; epilogue scales normalized v_diff, scatters 3 planes
        {
            dim3 grid(EE / 128, 1);
            size_t smem = (size_t)(128 * 192 + 128 * 128) * 4;
            painn_mlp_kernel<192, A_GATHER, E_SCATTER_V, true, false>
                <<<grid, threads, smem, stream>>>(
                    nullptr, rbf, s_cur, ei, ei + EE,
                    wp_b[b][2], F32(pb + 4), wp_b[b][3], F32(pb + 6),
                    v_cur, dv_agg, 128);
        }
        // upd_scalar: concat(s, ds_agg) [N,256] -> MLP2 -> + s
        {
            dim3 grid(NN / 128, 1);
            size_t smem = (size_t)(128 * 256 + 128 * 128) * 4;   // 192 KB
            painn_mlp_kernel<256, A_CONCAT2, E_RESIDUAL, true, false>
                <<<grid, threads, smem, stream>>>(
                    s_cur, ds_agg, nullptr, nullptr, nullptr,
                    wp_b[b][4], F32(pb + 8), wp_b[b][5], F32(pb + 10),
                    s_cur, s_nxt, 128);
        }
        // upd_vector: concat(v, dv_agg) [3N,256] -> MLP2 -> + v
        {
            dim3 grid(3 * NN / 128, 1);
            size_t smem = (size_t)(128 * 256 + 128 * 128) * 4;
            painn_mlp_kernel<256, A_CONCAT2, E_RESIDUAL, true, false>
                <<<grid, threads, smem, stream>>>(
                    v_cur, dv_agg, nullptr, nullptr, nullptr,
                    wp_b[b][6], F32(pb + 12), wp_b[b][7], F32(pb + 14),
                    v_cur, v_nxt, 128);
        }
        { float* t = s_cur; s_cur = s_nxt; s_nxt = t; }
        { float* t = v_cur; v_cur = v_nxt; v_nxt = t; }
    }

    // head: h = silu(s@out1+b1); h = silu(h@out2+b2)
    {
        dim3 grid(NN / 128, 1);
        size_t smem = (size_t)(128 * 128 + 128 * 128) * 4;       // 128 KB
        painn_mlp_kernel<128, A_PLAIN, E_STORE, true, true>
            <<<grid, threads, smem, stream>>>(
                s_cur, nullptr, nullptr, nullptr, nullptr,
                wp_out1, F32(55), wp_out2, F32(57),
                nullptr, hhead, 128);
    }
    // out = h @ w3 + b3
    out3_kernel<<<(NN + 255) / 256, threads, 0, stream>>>(
        hhead, F32(60), F32(59), (float*)d_out);
}